// GCNet_87522843558074
// MI455X (gfx1250) — compile-verified
//
#include <hip/hip_runtime.h>

typedef __attribute__((ext_vector_type(2))) float v2f;
typedef __attribute__((ext_vector_type(8))) float v8f;

// ---------------------------------------------------------------- utilities
__global__ void set_val_kernel(float* p, long long n, float v) {
  long long i = (long long)blockIdx.x * blockDim.x + threadIdx.x;
  if (i < n) p[i] = v;
}

__global__ void deg_count_kernel(const int* __restrict__ dst, int E, float* deg) {
  int e = blockIdx.x * blockDim.x + threadIdx.x;
  if (e < E) atomicAdd(&deg[dst[e]], 1.0f);
}

__global__ void rsqrt_kernel(float* p, int n) {
  int i = blockIdx.x * blockDim.x + threadIdx.x;
  if (i < n) p[i] = rsqrtf(p[i]);
}

// ---------------------------------------------------------------- WMMA GEMM
// OUT[M x N] = X[M x K] * W[K x N], all row-major, f32 via V_WMMA_F32_16X16X4_F32.
// 256 threads = 8 waves; each wave computes one 16x16 tile.
// A 16x4 layout: lane m = lane&15, regs hold K = 2*(lane>>4)+{0,1}  (float2 load)
// B 4x16 layout: lane n = lane&15, regs hold K = 2*(lane>>4)+{0,1}  (from LDS)
// C/D 16x16: reg r -> row r + 8*(lane>>4), col lane&15
template<int K, int N>
__global__ void gemm_wmma_kernel(const float* __restrict__ X,
                                 const float* __restrict__ W,
                                 float* __restrict__ OUT, int M) {
  __shared__ float sW[K * N];
  for (int i = threadIdx.x; i < K * N; i += blockDim.x) sW[i] = W[i];
  __syncthreads();

  constexpr int COL_TILES  = N / 16;       // 4 (N=64) or 2 (N=32)
  constexpr int ROW_TILES  = 8 / COL_TILES;
  constexpr int BLOCK_ROWS = 16 * ROW_TILES;

  const int wave = threadIdx.x >> 5;
  const int lane = threadIdx.x & 31;
  const int rowBase = blockIdx.x * BLOCK_ROWS + (wave / COL_TILES) * 16;
  const int colBase = (wave % COL_TILES) * 16;
  if (rowBase + 16 > M) return;            // wave-uniform: EXEC all-ones in WMMA

  const int mn  = lane & 15;               // A row / B col / D col
  const int grp = lane >> 4;               // K sub-group (0 or 1)

  const float* xrow = X + (long long)(rowBase + mn) * K + 2 * grp;
  const float* wcol = sW + (long long)(2 * grp) * N + colBase + mn;

  v8f acc = {};
#pragma unroll 4
  for (int k0 = 0; k0 < K; k0 += 4) {
    v2f a = *(const v2f*)(xrow + k0);      // X[row][k0+2g .. k0+2g+1]
    v2f b;
    b.x = wcol[(long long)k0 * N];         // W[k0+2g  ][col]
    b.y = wcol[(long long)(k0 + 1) * N];   // W[k0+2g+1][col]
    acc = __builtin_amdgcn_wmma_f32_16x16x4_f32(
        false, a, false, b, (short)0, acc, false, false);
  }

  float* out = OUT + (long long)(rowBase + 8 * grp) * N + colBase + mn;
#pragma unroll
  for (int r = 0; r < 8; ++r) out[(long long)r * N] = acc[r];
}

// ---------------------------------------------------------------- edge scatter
// thread t -> edge t/CH, 4-float chunk t%CH; vector gather + 4 f32 atomics.
template<int D>
__global__ void scatter_kernel(const int* __restrict__ src, const int* __restrict__ dst,
                               const float* __restrict__ dis, const float* __restrict__ H,
                               float* __restrict__ agg, int E) {
  constexpr int CH = D / 4;
  long long t = (long long)blockIdx.x * blockDim.x + threadIdx.x;
  long long e = t / CH;
  int c = (int)(t % CH);
  if (e >= E) return;
  int s = src[e], d = dst[e];
  float norm = dis[s] * dis[d];
  const float4 v = *(const float4*)(H + (long long)s * D + 4 * c);
  float* o = agg + (long long)d * D + 4 * c;
  atomicAdd(o + 0, v.x * norm);
  atomicAdd(o + 1, v.y * norm);
  atomicAdd(o + 2, v.z * norm);
  atomicAdd(o + 3, v.w * norm);
}

// ---------------------------------------------------------------- node combine
template<int D, bool RELU>
__global__ void combine_kernel(const float* __restrict__ H, const float* __restrict__ bias,
                               const float* __restrict__ dis, float* agg, int M) {
  long long i = (long long)blockIdx.x * blockDim.x + threadIdx.x;
  if (i >= (long long)M * D) return;
  int n = (int)(i / D), d = (int)(i % D);
  float z = agg[i] + dis[n] * dis[n] * H[i] + bias[d];
  if (RELU) z = fmaxf(z, 0.0f);
  agg[i] = z;
}

// combine + row softmax; one wave32 per node, one lane per output dim (D=32)
__global__ void combine_softmax_kernel(const float* __restrict__ H2,
                                       const float* __restrict__ b2,
                                       const float* __restrict__ dis,
                                       float* out, int M) {
  long long gt = (long long)blockIdx.x * blockDim.x + threadIdx.x;
  int node = (int)(gt >> 5);
  int lane = threadIdx.x & 31;
  if (node >= M) return;
  long long idx = (long long)node * 32 + lane;
  float ds2 = dis[node];
  ds2 *= ds2;
  float z = out[idx] + ds2 * H2[idx] + b2[lane];
  float m = z;
  for (int off = 16; off > 0; off >>= 1) m = fmaxf(m, __shfl_xor(m, off, 32));
  float ev = __expf(z - m);
  float s = ev;
  for (int off = 16; off > 0; off >>= 1) s += __shfl_xor(s, off, 32);
  out[idx] = ev / s;
}

// ---------------------------------------------------------------- launcher
extern "C" void kernel_launch(void* const* d_in, const int* in_sizes, int n_in,
                              void* d_out, int out_size, void* d_ws, size_t ws_size,
                              hipStream_t stream) {
  (void)n_in; (void)out_size; (void)ws_size;
  const float* x  = (const float*)d_in[0];   // [M,256]
  const int*   ei = (const int*)d_in[1];     // [2,E]
  const float* W1 = (const float*)d_in[2];   // [256,64]
  const float* b1 = (const float*)d_in[3];   // [64]
  const float* W2 = (const float*)d_in[4];   // [64,32]
  const float* b2 = (const float*)d_in[5];   // [32]

  const int IN = 256, HID = 64, OD = 32;
  const int M = in_sizes[0] / IN;   // 100000
  const int E = in_sizes[1] / 2;    // 1600000
  const int* src = ei;
  const int* dst = ei + E;

  float* ws   = (float*)d_ws;
  float* dis  = ws;                                   // M      (deg -> 1/sqrt(deg))
  float* H1   = dis + (((long long)M + 1023) & ~1023LL); // M*64 (x @ W1)
  float* agg1 = H1 + (long long)M * HID;              // M*64   (agg -> relu'd h1)
  float* H2   = agg1 + (long long)M * HID;            // M*32   (h1 @ W2)
  float* out  = (float*)d_out;                        // M*32   (agg2 -> softmax)

  const int T = 256;

  // degrees: deg = 1 + sum_edges(dst) ; dis = rsqrt(deg)
  set_val_kernel<<<(M + T - 1) / T, T, 0, stream>>>(dis, M, 1.0f);
  deg_count_kernel<<<(E + T - 1) / T, T, 0, stream>>>(dst, E, dis);
  rsqrt_kernel<<<(M + T - 1) / T, T, 0, stream>>>(dis, M);

  // ---- layer 1
  gemm_wmma_kernel<256, 64><<<(M + 31) / 32, 256, 0, stream>>>(x, W1, H1, M);
  set_val_kernel<<<(int)(((long long)M * HID + T - 1) / T), T, 0, stream>>>(
      agg1, (long long)M * HID, 0.0f);
  {
    long long tot = (long long)E * (HID / 4);
    scatter_kernel<64><<<(int)((tot + T - 1) / T), T, 0, stream>>>(src, dst, dis, H1, agg1, E);
  }
  combine_kernel<64, true><<<(int)(((long long)M * HID + T - 1) / T), T, 0, stream>>>(
      H1, b1, dis, agg1, M);

  // ---- layer 2
  gemm_wmma_kernel<64, 32><<<(M + 63) / 64, 256, 0, stream>>>(agg1, W2, H2, M);
  set_val_kernel<<<(int)(((long long)M * OD + T - 1) / T), T, 0, stream>>>(
      out, (long long)M * OD, 0.0f);
  {
    long long tot = (long long)E * (OD / 4);
    scatter_kernel<32><<<(int)((tot + T - 1) / T), T, 0, stream>>>(src, dst, dis, H2, out, E);
  }
  combine_softmax_kernel<<<(int)(((long long)M * 32 + T - 1) / T), T, 0, stream>>>(
      H2, b2, dis, out, M);
}